// SLUTagging_83863531422426
// MI455X (gfx1250) — compile-verified
//
#include <hip/hip_runtime.h>

typedef float v4f __attribute__((ext_vector_type(4)));

namespace {
constexpr int   kB = 256, kS = 512, kW = 512, kH = 512;
constexpr int   kRows = kB * kS;          // 131072 (b,s) rows
constexpr int   kRowsPerBlock = 8;        // 512 % 8 == 0 -> a block never crosses a batch
constexpr float kRateHead = 0.88f;
constexpr float kRateMid  = 0.70f;
}  // namespace

// Each 128-lane group processes one H=512 row as 128 x float4 (b128 VMEM ops).
// 256 threads/block = 8 wave32 = two row-groups; the block walks 8 consecutive
// rows so the sorted word_idx segments reuse the same out_word row from cache.
// All per-iteration char/out addresses are constant strides off one base
// pointer, so the unrolled loop folds them into immediate VMEM offsets.
__global__ __launch_bounds__(256) void slu_gather_blend(
    const float* __restrict__ out_word,   // [B, W, H]
    const float* __restrict__ out_char,   // [B, S, H]
    const int*   __restrict__ word_idx,   // [B, S] sorted per batch
    float* __restrict__ out)              // [B, S, H]
{
  const int  group = threadIdx.x >> 7;    // 0..1 : which row of each pair
  const int  lane  = threadIdx.x & 127;   // float4 slot within the H=512 row
  const long r0    = (long)blockIdx.x * kRowsPerBlock + group;  // first row
  const int  s0    = (int)(r0 & (kS - 1));
  const long b     = r0 >> 9;             // batch: identical for all 8 rows

  const float* wbase = out_word + b * (long)kW * kH + (long)lane * 4;
  const float* cp    = out_char + r0 * (long)kH + (long)lane * 4;
  float*       op    = out      + r0 * (long)kH + (long)lane * 4;
  const int*   ip    = word_idx + r0;

#pragma unroll
  for (int i = 0; i < kRowsPerBlock / 2; ++i) {
    const int s  = s0 + 2 * i;            // s==0 only possible at i==0, group==0
    const int wi = ip[2 * i];
    // Branch-free prev: clamp the -1 offset away at a batch head; the s==0
    // term below forces "head" there regardless of the (self-)loaded value.
    const int prev = ip[2 * i - (s != 0 ? 1 : 0)];
    const float rate = (s == 0 || wi != prev) ? kRateHead : kRateMid;
    const float inv  = 1.0f - rate;

    // gfx1250 prefetch path (global_prefetch_b8): next pair's char row.
    if (i + 1 < kRowsPerBlock / 2) {
      __builtin_prefetch(cp + (i + 1) * 2 * kH, 0, 0);
    }

    const v4f wv = *(const v4f*)(wbase + (long)wi * kH);               // cached
    const v4f cv = __builtin_nontemporal_load((const v4f*)(cp + i * 2 * kH));
    const v4f ov = wv * rate + cv * inv;                               // v_fma
    __builtin_nontemporal_store(ov, (v4f*)(op + i * 2 * kH));
  }
}

extern "C" void kernel_launch(void* const* d_in, const int* in_sizes, int n_in,
                              void* d_out, int out_size, void* d_ws, size_t ws_size,
                              hipStream_t stream) {
  (void)in_sizes; (void)n_in; (void)out_size; (void)d_ws; (void)ws_size;

  const float* out_word = (const float*)d_in[0];  // [B, W, H] f32
  const float* out_char = (const float*)d_in[1];  // [B, S, H] f32
  const int*   word_idx = (const int*)  d_in[2];  // [B, S]    i32
  float*       out      = (float*)d_out;          // [B, S, H] f32

  dim3 grid(kRows / kRowsPerBlock);  // 16384 blocks
  dim3 block(256);                   // 8 wave32
  hipLaunchKernelGGL(slu_gather_blend, grid, block, 0, stream,
                     out_word, out_char, word_idx, out);
}